// LGE_CB_71193377898902
// MI455X (gfx1250) — compile-verified
//
#include <hip/hip_runtime.h>
#include <hip/hip_bf16.h>

// ===========================================================================
// Lattice gauge-equivariant bilinear layer, fused, for MI455X (gfx1250).
//   out[o] = sum_n A_n * C[n,o],   C[n,o] = sum_m w[n,m,o] * B_m
// Stage A: fp64 transport T = U F(x+mu) U^dag  -> fp32 B panel in LDS
// Stage B: complex channel-mix GEMM via V_WMMA_F32_16X16X4_F32 (4 real WMMAs)
// Stage C: fp64 finish (site-local 3x3 complex matmuls) -> global
// Features_a staged via GLOBAL_LOAD_ASYNC_TO_LDS (ASYNCcnt) when available.
// ===========================================================================

#define L_DIM   16
#define V_SITES (L_DIM * L_DIM * L_DIM * L_DIM)   // 65536
#define NF      4
#define NF_OUT  4
#define SPB     8            // sites per block
#define NP      80           // padded N columns (8 sites * 9 entries = 72 -> 5 tiles)
#define KW      36           // padded K (33 B-channels -> 9 k-steps of 4)
#define MW      48           // padded M (36 (n,o) rows -> 3 tiles)
#define NTILE_M 3
#define NTILE_N 5
#define NTILE_K 9

typedef __attribute__((ext_vector_type(2))) float v2f;
typedef __attribute__((ext_vector_type(8))) float v8f;
typedef __attribute__((ext_vector_type(4))) int   v4i;
typedef __attribute__((address_space(1))) v4i     as1_v4i;   // global int4
typedef __attribute__((address_space(3))) v4i     as3_v4i;   // LDS int4

#if __has_builtin(__builtin_amdgcn_wmma_f32_16x16x4_f32)
#define HAVE_WMMA_F32X4 1
#endif
#if __has_builtin(__builtin_amdgcn_global_load_async_to_lds_b128)
#define HAVE_ASYNC_LDS 1
#endif

// ---- fp64 complex helpers --------------------------------------------------
__device__ __forceinline__ double2 cmadd(double2 a, double2 b, double2 c) {
  c.x = fma(a.x, b.x, fma(-a.y, b.y, c.x));
  c.y = fma(a.x, b.y, fma( a.y, b.x, c.y));
  return c;
}
__device__ __forceinline__ double2 cmaddCA(double2 a, double2 b, double2 c) { // conj(a)*b + c
  c.x = fma(a.x, b.x, fma( a.y, b.y, c.x));
  c.y = fma(a.x, b.y, fma(-a.y, b.x, c.y));
  return c;
}
__device__ __forceinline__ double2 cmaddCB(double2 a, double2 b, double2 c) { // a*conj(b) + c
  c.x = fma( a.x, b.x, fma(a.y, b.y, c.x));
  c.y = fma(-a.x, b.y, fma(a.y, b.x, c.y));
  return c;
}

// +1 hop (periodic) along mu; layout x = ((t*16+x)*16+y)*16+z, mu: 0=T..3=Z
__device__ __forceinline__ int neighbor_up(int x, int mu) {
  const int sh = (3 - mu) * 4;
  const int c  = (x >> sh) & (L_DIM - 1);
  return (x & ~((L_DIM - 1) << sh)) | (((c + 1) & (L_DIM - 1)) << sh);
}

// ---- weight pre-pack: w[9][33][4] c128 -> fp32 GEMM panels [MW][KW] --------
__global__ void lge_pack_weights(const double2* __restrict__ w,
                                 float* __restrict__ Wre,
                                 float* __restrict__ Wim,
                                 float* __restrict__ Wng) {
  const int idx = blockIdx.x * blockDim.x + threadIdx.x;
  if (idx >= MW * KW) return;
  const int M = idx / KW, K = idx % KW;      // M = n*4+o, K = m
  float re = 0.0f, im = 0.0f;
  if (M < 36 && K < 33) {
    const int n = M >> 2, o = M & 3;
    const double2 v = w[((size_t)n * 33 + K) * 4 + o];
    re = (float)v.x; im = (float)v.y;
  }
  Wre[idx] = re; Wim[idx] = im; Wng[idx] = -im;
}

// ---- fused main kernel -----------------------------------------------------
__global__ __launch_bounds__(128)
void lge_bilinear_kernel(const double2* __restrict__ Ug,
                         const double2* __restrict__ Fg,
                         const float*  __restrict__ Wre,
                         const float*  __restrict__ Wim,
                         const float*  __restrict__ Wng,
                         double2* __restrict__ outg) {
  __shared__ float   Bre[KW * NP];          // B panel (re), rows = m-channel
  __shared__ float   Bim[KW * NP];          // B panel (im)
  __shared__ float   Cre[MW * NP];          // C panel (re), rows = n*4+o
  __shared__ float   Cim[MW * NP];          // C panel (im)
  __shared__ double2 Fsh[SPB][NF][9];       // features_a operands, fp64

  const int t     = threadIdx.x;
  const int lane  = t & 31;
  const int wave  = t >> 5;
  const int site0 = blockIdx.x * SPB;

  // -- stage F(x) into LDS (async path; overlaps with the fp64 transport) ---
  if (t < SPB * NF) {
    const int s = t >> 2, f = t & 3;
    const int x = site0 + s;
    const double2* Fp = Fg + ((size_t)f * V_SITES + x) * 9;
#if defined(HAVE_ASYNC_LDS)
    const unsigned long long gsrc = (unsigned long long)(const void*)Fp;
    const unsigned int       ldst = (unsigned int)(unsigned long long)(const void*)&Fsh[s][f][0];
#pragma unroll
    for (int i = 0; i < 9; ++i) {           // 9 x 16B = one 3x3 c128 matrix
      __builtin_amdgcn_global_load_async_to_lds_b128(
          (as1_v4i*)(gsrc + 16ull * i), (as3_v4i*)(ldst + 16u * i), 0, 0);
    }
#else
#pragma unroll
    for (int i = 0; i < 9; ++i) Fsh[s][f][i] = Fp[i];
#endif
  }

  // -- prefetch this block's gauge links while the B panel is being zeroed --
  if (t < SPB * 4) {
    const int s = t >> 2, p = t & 3;
    __builtin_prefetch(&Ug[((size_t)p * V_SITES + site0 + s) * 9], 0, 1);
  }

  for (int i = t; i < KW * NP; i += 128) { Bre[i] = 0.0f; Bim[i] = 0.0f; }
  __syncthreads();

  // identity channel of features_b (m = 32)
  if (t < SPB) {
    const int cb = t * 9;
    Bre[32 * NP + cb + 0] = 1.0f;
    Bre[32 * NP + cb + 4] = 1.0f;
    Bre[32 * NP + cb + 8] = 1.0f;
  }

  // -- Stage A: fp64 transport, one (site, path, feature) task per thread ---
  {
    const int s = t >> 4, pf = t & 15, p = pf >> 2, f = pf & 3;
    const int x  = site0 + s;
    const int xn = neighbor_up(x, p);
    const double2* Up = Ug + ((size_t)p * V_SITES + x ) * 9;
    const double2* Fp = Fg + ((size_t)f * V_SITES + xn) * 9;
    double2 Um[9], Fn[9], M1[9];
#pragma unroll
    for (int i = 0; i < 9; ++i) { Um[i] = Up[i]; Fn[i] = Fp[i]; }
#pragma unroll
    for (int i = 0; i < 3; ++i)
#pragma unroll
      for (int k = 0; k < 3; ++k) {
        double2 acc = make_double2(0.0, 0.0);
#pragma unroll
        for (int j = 0; j < 3; ++j) acc = cmadd(Um[i*3+j], Fn[j*3+k], acc);
        M1[i*3+k] = acc;
      }
    const int cb = s * 9;
    const int m  = pf;                      // path-major, feature-minor
    const int ma = 16 + pf;                 // adjoint channel
#pragma unroll
    for (int i = 0; i < 3; ++i)
#pragma unroll
      for (int l = 0; l < 3; ++l) {
        double2 tv = make_double2(0.0, 0.0);
#pragma unroll
        for (int k = 0; k < 3; ++k) tv = cmaddCB(M1[i*3+k], Um[l*3+k], tv);
        Bre[m  * NP + cb + i*3 + l] =  (float)tv.x;
        Bim[m  * NP + cb + i*3 + l] =  (float)tv.y;
        Bre[ma * NP + cb + l*3 + i] =  (float)tv.x;   // conj-transpose
        Bim[ma * NP + cb + l*3 + i] = -(float)tv.y;
      }
  }

#if defined(HAVE_ASYNC_LDS)
#if __has_builtin(__builtin_amdgcn_s_wait_asynccnt)
  __builtin_amdgcn_s_wait_asynccnt(0);
#else
  asm volatile("s_wait_asynccnt 0" ::: "memory");
#endif
#endif
  __syncthreads();

  // -- Stage B: complex channel-mix GEMM via f32 WMMA 16x16x4 ---------------
  // Cr = Wr*Br + (-Wi)*Bi ; Ci = Wr*Bi + Wi*Br.  EXEC all-ones: the pair loop
  // is wave-uniform and unguarded across all 128 threads.
  for (int pair = wave; pair < NTILE_M * NTILE_N; pair += 4) {
    const int mt    = pair / NTILE_N, nt = pair % NTILE_N;
    const int arow  = mt * 16 + (lane & 15);      // A: M on lanes, K split hi/lo
    const int acol0 = (lane >> 4) << 1;           // lanes 0-15 -> K{0,1}, 16-31 -> K{2,3}
    const int bcol  = nt * 16 + (lane & 15);      // B: N on lanes
    v8f accr = {}; v8f acci = {};
#if defined(HAVE_WMMA_F32X4)
#pragma unroll
    for (int kt = 0; kt < NTILE_K; ++kt) {
      const int k0 = kt * 4 + acol0;
      v2f awr = { Wre[arow * KW + k0], Wre[arow * KW + k0 + 1] };
      v2f awi = { Wim[arow * KW + k0], Wim[arow * KW + k0 + 1] };
      v2f awn = { Wng[arow * KW + k0], Wng[arow * KW + k0 + 1] };
      v2f bbr = { Bre[k0 * NP + bcol], Bre[(k0 + 1) * NP + bcol] };
      v2f bbi = { Bim[k0 * NP + bcol], Bim[(k0 + 1) * NP + bcol] };
      accr = __builtin_amdgcn_wmma_f32_16x16x4_f32(false, awr, false, bbr, (short)0, accr, false, false);
      accr = __builtin_amdgcn_wmma_f32_16x16x4_f32(false, awn, false, bbi, (short)0, accr, false, false);
      acci = __builtin_amdgcn_wmma_f32_16x16x4_f32(false, awr, false, bbi, (short)0, acci, false, false);
      acci = __builtin_amdgcn_wmma_f32_16x16x4_f32(false, awi, false, bbr, (short)0, acci, false, false);
    }
#else
    // scalar fallback so the file still compiles if the builtin is absent
#pragma unroll
    for (int v = 0; v < 8; ++v) {
      const int Mr = mt * 16 + ((lane >> 4) << 3) + v;
      float sr = 0.0f, si = 0.0f;
      for (int k = 0; k < KW; ++k) {
        const float wr = Wre[Mr * KW + k], wi = Wim[Mr * KW + k];
        const float br = Bre[k * NP + bcol], bi = Bim[k * NP + bcol];
        sr = fmaf(wr, br, fmaf(-wi, bi, sr));
        si = fmaf(wr, bi, fmaf( wi, br, si));
      }
      accr[v] = sr; acci[v] = si;
    }
#endif
    const int mrow0 = mt * 16 + ((lane >> 4) << 3);   // D: VGPR v -> M=v / M=v+8
#pragma unroll
    for (int v = 0; v < 8; ++v) {
      Cre[(mrow0 + v) * NP + bcol] = accr[v];
      Cim[(mrow0 + v) * NP + bcol] = acci[v];
    }
  }
  __syncthreads();

  // -- Stage C: fp64 finish  out[o] = C[8,o] + sum_f F_f C[f,o] + F_f^† C[4+f,o]
  if (t < SPB * NF_OUT) {
    const int s = t >> 2, o = t & 3;
    const int x  = site0 + s;
    const int cb = s * 9;
    double2 acc[9];
    const int M8 = 8 * 4 + o;                         // identity A-channel
#pragma unroll
    for (int e = 0; e < 9; ++e)
      acc[e] = make_double2((double)Cre[M8 * NP + cb + e],
                            (double)Cim[M8 * NP + cb + e]);
#pragma unroll
    for (int f = 0; f < NF; ++f) {
      const int Md = f * 4 + o;
      const int Ma = 16 + f * 4 + o;
#pragma unroll
      for (int i = 0; i < 3; ++i)
#pragma unroll
        for (int k = 0; k < 3; ++k) {
          double2 a = acc[i*3+k];
#pragma unroll
          for (int j = 0; j < 3; ++j) {
            const double2 Aij = Fsh[s][f][i*3+j];
            const double2 c1  = make_double2((double)Cre[Md * NP + cb + j*3 + k],
                                             (double)Cim[Md * NP + cb + j*3 + k]);
            a = cmadd(Aij, c1, a);
            const double2 Aji = Fsh[s][f][j*3+i];     // conj(A[j][i])
            const double2 c2  = make_double2((double)Cre[Ma * NP + cb + j*3 + k],
                                             (double)Cim[Ma * NP + cb + j*3 + k]);
            a = cmaddCA(Aji, c2, a);
          }
          acc[i*3+k] = a;
        }
    }
    double2* op = outg + ((size_t)o * V_SITES + x) * 9;
#pragma unroll
    for (int e = 0; e < 9; ++e) op[e] = acc[e];
  }
}

// ---------------------------------------------------------------------------
extern "C" void kernel_launch(void* const* d_in, const int* in_sizes, int n_in,
                              void* d_out, int out_size, void* d_ws, size_t ws_size,
                              hipStream_t stream) {
  const double2* U = (const double2*)d_in[0];   // [4][V][3][3] complex128
  const double2* F = (const double2*)d_in[1];   // [4][V][3][3] complex128
  const double2* W = (const double2*)d_in[2];   // [9][33][4]   complex128
  float* Wre = (float*)d_ws;
  float* Wim = Wre + MW * KW;
  float* Wng = Wim + MW * KW;                   // total 3*48*36*4 = 20736 B of ws
  double2* out = (double2*)d_out;

  lge_pack_weights<<<(MW * KW + 127) / 128, 128, 0, stream>>>(W, Wre, Wim, Wng);
  lge_bilinear_kernel<<<V_SITES / SPB, 128, 0, stream>>>(U, F, Wre, Wim, Wng, out);

  (void)in_sizes; (void)n_in; (void)out_size; (void)ws_size;
}